// GAT_20942260536008
// MI455X (gfx1250) — compile-verified
//
#include <hip/hip_runtime.h>

#define N_NODES 50000
#define N_EDGES 800000
#define TOTC    128          // hidden width every layer (H*C = 128, OUT = 128)
#define NCLS    10
#define NGRAPH  64
#define SLOPE   0.2f
#define BN_EPS  1e-5f

typedef float v2f __attribute__((ext_vector_type(2)));
typedef float v8f __attribute__((ext_vector_type(8)));

// ---------------- fills (graph-capture-safe init) ----------------
__global__ __launch_bounds__(256) void k_fill_f32(float* p, long long n, float v) {
  long long i = blockIdx.x * 256LL + threadIdx.x;
  if (i < n) p[i] = v;
}
__global__ __launch_bounds__(256) void k_fill_u32(unsigned* p, long long n, unsigned v) {
  long long i = blockIdx.x * 256LL + threadIdx.x;
  if (i < n) p[i] = v;
}

// ---------------- fp32 WMMA GEMM: Hout[N,128] = X[N,128] @ W[128,128] ----------------
// One wave per 16x16 output tile; block = 8 waves = one 16-row stripe of all 128 cols.
// K-loop: 32 x v_wmma_f32_16x16x4_f32, fully unrolled.
__global__ __launch_bounds__(256) void k_gemm_wmma(const float* __restrict__ X,
                                                   const float* __restrict__ W,
                                                   float* __restrict__ Hout) {
  const int lane = threadIdx.x & 31;
  const int wv   = threadIdx.x >> 5;     // col tile 0..7
  const int row0 = blockIdx.x << 4;      // 16 rows / block (N % 16 == 0)
  const int col0 = wv << 4;
  const int half = lane >> 4;            // 0: K={k,k+1}   1: K={k+2,k+3}
  const int lr   = lane & 15;            // A-row / B-col within tile

  const float* __restrict__ xrow = X + (size_t)(row0 + lr) * TOTC;
  const float* __restrict__ wcol = W + col0 + lr;

  v8f acc = {0.f, 0.f, 0.f, 0.f, 0.f, 0.f, 0.f, 0.f};
#pragma unroll
  for (int k = 0; k < TOTC; k += 4) {
    const int ka = k + (half << 1);
    v2f a = *(const v2f*)(xrow + ka);                 // A 16x4: VGPR0=K even-pair lo, VGPR1=+1
    v2f b;
    b.x = wcol[(size_t)ka * TOTC];                    // B 4x16: row K=ka across lanes
    b.y = wcol[(size_t)(ka + 1) * TOTC];
    acc = __builtin_amdgcn_wmma_f32_16x16x4_f32(false, a, false, b,
                                                (short)0, acc, false, false);
  }
  // D 16x16 f32: VGPR v -> M=v (lanes 0-15) / M=v+8 (lanes 16-31), N = lane%16
  float* __restrict__ orow = Hout + (size_t)(row0 + (half << 3)) * TOTC + col0 + lr;
#pragma unroll
  for (int v = 0; v < 8; ++v)
    orow[(size_t)v * TOTC] = acc[v];
}

// ---------------- attention logits: al_s/al_d [N,H] ----------------
__global__ __launch_bounds__(256) void k_attn(const float* __restrict__ Hb,
                                              const float* __restrict__ asv,
                                              const float* __restrict__ adv,
                                              float* __restrict__ als,
                                              float* __restrict__ ald,
                                              int Hh, int hshift, int Cc) {
  long long gid = blockIdx.x * 256LL + threadIdx.x;
  if (gid >= (long long)N_NODES * Hh) return;
  int n  = (int)(gid >> hshift);
  int hh = (int)(gid & (Hh - 1));
  const float* hr = Hb + (size_t)n * TOTC + hh * Cc;
  const float* sa = asv + hh * Cc;
  const float* da = adv + hh * Cc;
  float accs = 0.f, accd = 0.f;
  for (int c = 0; c < Cc; c += 4) {
    float4 hv = *(const float4*)(hr + c);
    float4 sv = *(const float4*)(sa + c);
    float4 dv = *(const float4*)(da + c);
    accs += hv.x * sv.x + hv.y * sv.y + hv.z * sv.z + hv.w * sv.w;
    accd += hv.x * dv.x + hv.y * dv.y + hv.z * dv.z + hv.w * dv.w;
  }
  als[gid] = accs;
  ald[gid] = accd;
}

// monotonic float<->uint for atomicMax-based segment max
__device__ __forceinline__ unsigned fenc(float f) {
  unsigned u = __float_as_uint(f);
  return (u & 0x80000000u) ? ~u : (u | 0x80000000u);
}
__device__ __forceinline__ float fdec(unsigned e) {
  unsigned u = (e & 0x80000000u) ? (e & 0x7FFFFFFFu) : ~e;
  return __uint_as_float(u);
}

// ---------------- pass 1: alpha = leakyrelu(al_s[s]+al_d[d]); segment max over dst ----------------
__global__ __launch_bounds__(256) void k_alpha_max(const float* __restrict__ als,
                                                   const float* __restrict__ ald,
                                                   const int* __restrict__ src,
                                                   const int* __restrict__ dst,
                                                   float* __restrict__ aArr,
                                                   unsigned* __restrict__ mEnc,
                                                   int Hh, int hshift) {
  long long gid = blockIdx.x * 256LL + threadIdx.x;
  long long total = (long long)(N_EDGES + N_NODES) * Hh;
  if (gid >= total) return;
  long long ei = gid >> hshift;
  int hh = (int)(gid & (Hh - 1));
  int s, d;
  if (ei < N_EDGES) { s = src[ei]; d = dst[ei]; }
  else              { s = d = (int)(ei - N_EDGES); }   // self-loops
  float a = als[(size_t)s * Hh + hh] + ald[(size_t)d * Hh + hh];
  a = (a > 0.f) ? a : a * SLOPE;
  aArr[gid] = a;
  atomicMax(mEnc + (size_t)d * Hh + hh, fenc(a));
}

// ---------------- pass 2: e = exp(alpha - m[d]); segment sum over dst ----------------
__global__ __launch_bounds__(256) void k_expsum(const int* __restrict__ dst,
                                                float* __restrict__ aArr,
                                                const unsigned* __restrict__ mEnc,
                                                float* __restrict__ denom,
                                                int Hh, int hshift) {
  long long gid = blockIdx.x * 256LL + threadIdx.x;
  long long total = (long long)(N_EDGES + N_NODES) * Hh;
  if (gid >= total) return;
  long long ei = gid >> hshift;
  int hh = (int)(gid & (Hh - 1));
  int d = (ei < N_EDGES) ? dst[ei] : (int)(ei - N_EDGES);
  float m = fdec(mEnc[(size_t)d * Hh + hh]);
  float e = __expf(aArr[gid] - m);
  aArr[gid] = e;                                    // reuse: stores e for pass 3
  atomicAdd(denom + (size_t)d * Hh + hh, e);
}

// ---------------- pass 3: acc[d] += h[s] * (e/denom[d]) ; one wave per edge, 4 ch/lane ----------------
__global__ __launch_bounds__(256) void k_aggregate(const float* __restrict__ Hb,
                                                   const float* __restrict__ eArr,
                                                   const float* __restrict__ denom,
                                                   const int* __restrict__ src,
                                                   const int* __restrict__ dst,
                                                   float* __restrict__ acc,
                                                   int Hh, int cshift) {
  long long gid = blockIdx.x * 256LL + threadIdx.x;
  long long ei  = gid >> 5;
  if (ei >= (long long)(N_EDGES + N_NODES)) return;
  int lane = (int)(gid & 31);
  int s, d;
  if (ei < N_EDGES) { s = src[ei]; d = dst[ei]; }
  else              { s = d = (int)(ei - N_EDGES); }
  int c4   = lane << 2;                 // channel start: 0..124
  int head = c4 >> cshift;              // C=32 -> >>5 ; C=128 -> >>7
  float coef = eArr[ei * Hh + head] / denom[(size_t)d * Hh + head];
  float4 hv = *(const float4*)(Hb + (size_t)s * TOTC + c4);
  float* o = acc + (size_t)d * TOTC + c4;
  atomicAdd(o + 0, hv.x * coef);
  atomicAdd(o + 1, hv.y * coef);
  atomicAdd(o + 2, hv.z * coef);
  atomicAdd(o + 3, hv.w * coef);
}

// ---------------- bias + BatchNorm(eval) + ReLU (in place OK) ----------------
__global__ __launch_bounds__(256) void k_bias_bn_relu(const float* __restrict__ acc,
                                                      const float* __restrict__ bias,
                                                      const float* __restrict__ g,
                                                      const float* __restrict__ bb,
                                                      const float* __restrict__ rm,
                                                      const float* __restrict__ rv,
                                                      float* __restrict__ out) {
  long long gid = blockIdx.x * 256LL + threadIdx.x;
  if (gid >= (long long)N_NODES * TOTC) return;
  int ch = (int)(gid & (TOTC - 1));
  float v = acc[gid] + bias[ch];
  v = g[ch] * (v - rm[ch]) * __frsqrt_rn(rv[ch] + BN_EPS) + bb[ch];
  out[gid] = fmaxf(v, 0.f);
}

// ---------------- global add pool ----------------
__global__ __launch_bounds__(256) void k_pool(const float* __restrict__ x,
                                              const int* __restrict__ batch,
                                              float* __restrict__ pooled) {
  long long gid = blockIdx.x * 256LL + threadIdx.x;
  long long n = gid >> 5;
  if (n >= N_NODES) return;
  int lane = (int)(gid & 31);
  int c4 = lane << 2;
  int g = batch[n];
  float4 v = *(const float4*)(x + (size_t)n * TOTC + c4);
  float* o = pooled + (size_t)g * TOTC + c4;
  atomicAdd(o + 0, v.x);
  atomicAdd(o + 1, v.y);
  atomicAdd(o + 2, v.z);
  atomicAdd(o + 3, v.w);
}

// ---------------- classifier: [64,128] @ [128,10] + bc ----------------
__global__ __launch_bounds__(1024) void k_cls(const float* __restrict__ pooled,
                                              const float* __restrict__ Wc,
                                              const float* __restrict__ bc,
                                              float* __restrict__ out) {
  int t = threadIdx.x;
  if (t >= NGRAPH * NCLS) return;
  int g = t / NCLS, c = t % NCLS;
  float s = bc[c];
  for (int k = 0; k < TOTC; ++k)
    s += pooled[(size_t)g * TOTC + k] * Wc[(size_t)k * NCLS + c];
  out[t] = s;
}

static inline int cdiv(long long a, long long b) { return (int)((a + b - 1) / b); }

extern "C" void kernel_launch(void* const* d_in, const int* in_sizes, int n_in,
                              void* d_out, int out_size, void* d_ws, size_t ws_size,
                              hipStream_t stream) {
  (void)in_sizes; (void)n_in; (void)out_size; (void)ws_size;

  const float* x     = (const float*)d_in[0];
  const int*   eidx  = (const int*)d_in[1];
  const int*   src   = eidx;
  const int*   dst   = eidx + N_EDGES;
  const int*   batch = (const int*)d_in[2];
  // per-layer params: base 3 + 8*l : W, as, ad, b, g, bb, rm, rv
  const float* Wl[3], *ASl[3], *ADl[3], *Bl[3], *Gl[3], *BBl[3], *RMl[3], *RVl[3];
  for (int l = 0; l < 3; ++l) {
    int base = 3 + 8 * l;
    Wl[l]  = (const float*)d_in[base + 0];
    ASl[l] = (const float*)d_in[base + 1];
    ADl[l] = (const float*)d_in[base + 2];
    Bl[l]  = (const float*)d_in[base + 3];
    Gl[l]  = (const float*)d_in[base + 4];
    BBl[l] = (const float*)d_in[base + 5];
    RMl[l] = (const float*)d_in[base + 6];
    RVl[l] = (const float*)d_in[base + 7];
  }
  const float* Wc = (const float*)d_in[27];
  const float* bc = (const float*)d_in[28];

  // workspace carve-up (floats)
  float* ws     = (float*)d_ws;
  float* hbuf   = ws;                                        // [N,128] GEMM result h
  float* bufA   = hbuf + (size_t)N_NODES * TOTC;             // [N,128]
  float* bufB   = bufA + (size_t)N_NODES * TOTC;             // [N,128]
  float* als    = bufB + (size_t)N_NODES * TOTC;             // [N,4]
  float* ald    = als + (size_t)N_NODES * 4;                 // [N,4]
  unsigned* mEnc= (unsigned*)(ald + (size_t)N_NODES * 4);    // [N,4]
  float* denom  = (float*)mEnc + (size_t)N_NODES * 4;        // [N,4]
  float* eArr   = denom + (size_t)N_NODES * 4;               // [(E+N),4]
  float* pooled = eArr + (size_t)(N_EDGES + N_NODES) * 4;    // [64,128]

  const long long nNode128 = (long long)N_NODES * TOTC;

  auto run_layer = [&](const float* xin, float* accout, int l,
                       int Hh, int hshift, int Cc, int cshift) {
    // 1) h = x @ W  (fp32 WMMA)
    k_gemm_wmma<<<N_NODES / 16, 256, 0, stream>>>(xin, Wl[l], hbuf);
    // 2) attention logits
    long long na = (long long)N_NODES * Hh;
    k_attn<<<cdiv(na, 256), 256, 0, stream>>>(hbuf, ASl[l], ADl[l], als, ald, Hh, hshift, Cc);
    // 3) init segment state + accumulator
    k_fill_u32<<<cdiv(na, 256), 256, 0, stream>>>(mEnc, na, 0u);      // enc(-inf-ish)
    k_fill_f32<<<cdiv(na, 256), 256, 0, stream>>>(denom, na, 0.f);
    k_fill_f32<<<cdiv(nNode128, 256), 256, 0, stream>>>(accout, nNode128, 0.f);
    // 4) segment softmax over dst + message aggregation
    long long ne = (long long)(N_EDGES + N_NODES) * Hh;
    k_alpha_max<<<cdiv(ne, 256), 256, 0, stream>>>(als, ald, src, dst, eArr, mEnc, Hh, hshift);
    k_expsum<<<cdiv(ne, 256), 256, 0, stream>>>(dst, eArr, mEnc, denom, Hh, hshift);
    long long nagg = (long long)(N_EDGES + N_NODES) * 32;
    k_aggregate<<<cdiv(nagg, 256), 256, 0, stream>>>(hbuf, eArr, denom, src, dst, accout, Hh, cshift);
    // 5) bias + BN + ReLU (in place)
    k_bias_bn_relu<<<cdiv(nNode128, 256), 256, 0, stream>>>(accout, Bl[l], Gl[l], BBl[l],
                                                            RMl[l], RVl[l], accout);
  };

  run_layer(x,    bufA, 0, 4, 2, 32, 5);
  run_layer(bufA, bufB, 1, 4, 2, 32, 5);
  run_layer(bufB, bufA, 2, 1, 0, 128, 7);

  // global add pool + classifier
  k_fill_f32<<<cdiv((long long)NGRAPH * TOTC, 256), 256, 0, stream>>>(pooled,
                                                (long long)NGRAPH * TOTC, 0.f);
  k_pool<<<cdiv((long long)N_NODES * 32, 256), 256, 0, stream>>>(bufA, batch, pooled);
  k_cls<<<1, 1024, 0, stream>>>(pooled, Wc, bc, (float*)d_out);
}